// SecondaryCapsules_39900246180128
// MI455X (gfx1250) — compile-verified
//
#include <hip/hip_runtime.h>
#include <math.h>

typedef __attribute__((ext_vector_type(2))) float v2f;
typedef __attribute__((ext_vector_type(8))) float v8f;

#define BATCH_N   512
#define NROUTES   512
#define CIN       8
#define COUT      16
#define NCAPS     10

#define XS_STRIDE 9    // padded row stride for x tile (bank-conflict-free A fetch)
#define U_STRIDE  17   // padded row stride for u_hat
#define NTHREADS  256  // 8 waves of 32

__launch_bounds__(NTHREADS)
__global__ void caps_route_kernel(const float* __restrict__ x,
                                  const float* __restrict__ rw,
                                  float* __restrict__ out)
{
    // LDS budget: 18432 + 512 + 34816 + 2048 + 2048 + 1024 + 64 + 32 + 8 ~= 57.6 KB
    __shared__ float xs[NROUTES * XS_STRIDE];   // x[bat] 512x8 (padded)
    __shared__ float Ws[CIN * COUT];            // route_weights[cap][bat] 8x16
    __shared__ float u_lds[NROUTES * U_STRIDE]; // u_hat 512x16 (padded)
    __shared__ float bl[NROUTES];               // routing logits b_ij
    __shared__ float el[NROUTES];               // exp(b - max)
    __shared__ float red[16 * 16];              // partial sums for s_j
    __shared__ float sarr[COUT];                // s_j (pre-squash)
    __shared__ float wred[8];                   // per-wave reduction slots
    __shared__ float s_scalar[2];               // [0]=max/sum, [1]=squash scale

    const int tid  = threadIdx.x;
    const int lane = tid & 31;
    const int wave = tid >> 5;
    const int half = lane >> 4;   // 0: lanes 0-15, 1: lanes 16-31
    const int lm   = lane & 15;   // M row (A) / N col (B, D)

    const int cap = blockIdx.x;   // 0..9
    const int bat = blockIdx.y;   // 0..511

    // ---- stage x[bat] (512x8 f32) into LDS, coalesced float4 loads ----
    const float* xp = x + (size_t)bat * NROUTES * CIN;
    #pragma unroll
    for (int c = 0; c < 4; ++c) {
        int e = c * (NTHREADS * 4) + tid * 4;      // flat element idx, 16B aligned
        float4 v = *(const float4*)(xp + e);
        int r = e >> 3;                            // route
        int k = e & 7;                             // 0 or 4
        float* dst = &xs[r * XS_STRIDE + k];
        dst[0] = v.x; dst[1] = v.y; dst[2] = v.z; dst[3] = v.w;
    }
    // ---- W slice: reference broadcasting picks route_weights[cap][bat] ----
    const float* wp = rw + ((size_t)cap * NROUTES + bat) * (CIN * COUT);
    if (tid < CIN * COUT) Ws[tid] = wp[tid];
    // init logits to zero
    bl[tid] = 0.0f;
    bl[tid + 256] = 0.0f;
    __syncthreads();

    // ---- u_hat = x[bat] (512x8) @ W (8x16) via V_WMMA_F32_16X16X4_F32 ----
    // Each wave owns 4 tiles of 16 routes; 2 WMMAs per tile (K = 8 = 2*4).
    // A layout: lanes 0-15 -> M=lm, VGPR0=K0,VGPR1=K1; lanes 16-31 -> K2,K3.
    // B layout: VGPR0: row K0 (lanes 0-15) / K2 (16-31); VGPR1: K1 / K3.
    // D layout: VGPR j -> (M=j, N=lm) lanes 0-15, (M=j+8, N=lm) lanes 16-31.
    #pragma unroll
    for (int i = 0; i < 4; ++i) {
        const int r0 = (wave * 4 + i) * 16;
        v8f acc = {0.f, 0.f, 0.f, 0.f, 0.f, 0.f, 0.f, 0.f};
        #pragma unroll
        for (int k0 = 0; k0 < CIN; k0 += 4) {
            const int ka = k0 + half * 2;
            v2f a, b;
            a.x = xs[(r0 + lm) * XS_STRIDE + ka];
            a.y = xs[(r0 + lm) * XS_STRIDE + ka + 1];
            b.x = Ws[ka * COUT + lm];
            b.y = Ws[(ka + 1) * COUT + lm];
            acc = __builtin_amdgcn_wmma_f32_16x16x4_f32(
                      false, a, false, b, (short)0, acc, false, false);
        }
        #pragma unroll
        for (int j = 0; j < 8; ++j)
            u_lds[(r0 + j + half * 8) * U_STRIDE + lm] = acc[j];
    }
    __syncthreads();

    // ---- 3 iterations of dynamic routing, entirely in LDS/registers ----
    for (int it = 0; it < 3; ++it) {
        // softmax max over 512 logits
        float m = fmaxf(bl[tid], bl[tid + 256]);
        #pragma unroll
        for (int off = 16; off > 0; off >>= 1)
            m = fmaxf(m, __shfl_xor(m, off, 32));
        if (lane == 0) wred[wave] = m;
        __syncthreads();
        if (tid == 0) {
            float mm = wred[0];
            #pragma unroll
            for (int w = 1; w < 8; ++w) mm = fmaxf(mm, wred[w]);
            s_scalar[0] = mm;
        }
        __syncthreads();
        const float bmax = s_scalar[0];
        float e0 = expf(bl[tid] - bmax);
        float e1 = expf(bl[tid + 256] - bmax);
        el[tid] = e0;
        el[tid + 256] = e1;
        float se = e0 + e1;
        #pragma unroll
        for (int off = 16; off > 0; off >>= 1)
            se += __shfl_xor(se, off, 32);
        if (lane == 0) wred[wave] = se;
        __syncthreads();
        if (tid == 0) {
            float ss = 0.f;
            #pragma unroll
            for (int w = 0; w < 8; ++w) ss += wred[w];
            s_scalar[0] = ss;
        }
        __syncthreads();
        const float inv_sum = 1.0f / s_scalar[0];

        // s[o] = (1/sum) * sum_r exp(b[r]) * u_hat[r][o]
        // thread -> (route-group g = tid/16, out-channel o = tid%16);
        // each u_hat element is read exactly once per iteration.
        const int o = tid & 15;
        const int g = tid >> 4;
        float part = 0.f;
        #pragma unroll 4
        for (int r = g; r < NROUTES; r += 16)
            part += el[r] * u_lds[r * U_STRIDE + o];
        red[g * 16 + o] = part;
        __syncthreads();
        if (tid < 16) {
            float s = 0.f;
            #pragma unroll
            for (int gg = 0; gg < 16; ++gg) s += red[gg * 16 + tid];
            sarr[tid] = s * inv_sum;
        }
        __syncthreads();
        if (tid == 0) {
            float nsq = 0.f;
            #pragma unroll
            for (int oo = 0; oo < COUT; ++oo) nsq += sarr[oo] * sarr[oo];
            // squash: v = s * |s| / (1 + |s|^2)  ==  s * nsq / ((1+nsq)*sqrt(nsq))
            s_scalar[1] = nsq / ((1.0f + nsq) * sqrtf(nsq));
        }
        __syncthreads();

        if (it < 2) {  // final v doesn't depend on the last logit update
            const float sc = s_scalar[1];
            #pragma unroll
            for (int rr = 0; rr < 2; ++rr) {
                const int r = tid + rr * 256;
                const float* up = &u_lds[r * U_STRIDE];
                float dot = 0.f;
                #pragma unroll
                for (int oo = 0; oo < COUT; ++oo) dot += up[oo] * sarr[oo];
                bl[r] += dot * sc;
            }
            __syncthreads();
        }
    }

    // ---- output v_j: (BATCH, NCAPS, COUT) ----
    if (tid < COUT)
        out[((size_t)bat * NCAPS + cap) * COUT + tid] = sarr[tid] * s_scalar[1];
}

extern "C" void kernel_launch(void* const* d_in, const int* in_sizes, int n_in,
                              void* d_out, int out_size, void* d_ws, size_t ws_size,
                              hipStream_t stream) {
    (void)in_sizes; (void)n_in; (void)out_size; (void)d_ws; (void)ws_size;
    const float* x  = (const float*)d_in[0];
    const float* rw = (const float*)d_in[1];
    float* out = (float*)d_out;
    dim3 grid(NCAPS, BATCH_N);
    caps_route_kernel<<<grid, NTHREADS, 0, stream>>>(x, rw, out);
}